// HeterogeneousRGAT_88261577933334
// MI455X (gfx1250) — compile-verified
//
#include <hip/hip_runtime.h>
#include <math.h>

#define NEG_SLOPE 0.01f

typedef __attribute__((ext_vector_type(2))) float v2f;
typedef __attribute__((ext_vector_type(8))) float v8f;
typedef __attribute__((ext_vector_type(4))) int v4i;

// ---------------------------------------------------------------------------
// f32 WMMA wrapper: D = A(16x4) * B(4x16) + C(16x16), fp32 exact.
// ---------------------------------------------------------------------------
static __device__ inline v8f wmma_f32(v2f a, v2f b, v8f c) {
  return __builtin_amdgcn_wmma_f32_16x16x4_f32(false, a, false, b, (short)0, c,
                                               false, false);
}

// ---------------------------------------------------------------------------
// CDNA5 async global->LDS path (ASYNCcnt-tracked, ISA ch.08).
// Probe round 2 confirmed the builtin exists; param 1 is AS(1) int4*.
// Guarded: falls back to register staging if the builtins are absent.
// ---------------------------------------------------------------------------
#if defined(__HIP_DEVICE_COMPILE__) &&                                   \
    __has_builtin(__builtin_amdgcn_global_load_async_to_lds_b128) &&     \
    __has_builtin(__builtin_amdgcn_s_wait_asynccnt)
#define USE_ASYNC_LDS 1
#else
#define USE_ASYNC_LDS 0
#endif

#if USE_ASYNC_LDS
typedef __attribute__((address_space(1))) v4i* gptr_v4i;
typedef __attribute__((address_space(3))) v4i* lptr_v4i;
static __device__ inline void async_copy16(const float* gp, float* lp) {
  __builtin_amdgcn_global_load_async_to_lds_b128(
      (gptr_v4i)(void*)const_cast<float*>(gp), (lptr_v4i)(void*)lp, 0, 0);
}
static __device__ inline void async_wait0() {
  __builtin_amdgcn_s_wait_asynccnt(0);
}
#else
static __device__ inline void async_wait0() {}
#endif

// ---------------------------------------------------------------------------
// GEMM: C[M x Nc] = A[M x K] * Bm[K x Nc] (+ bias[Nc] if bias != nullptr)
// If attnp != nullptr, also accumulates (atomically):
//   ssrc[m] += C_row[m] . attnp[0:Nc] ; sdst[m] += C_row[m] . attnp[Nc:2Nc]
// Block: 256 threads (8 waves), tile 128x128, K-step 16, double-buffered LDS.
// Wave layout: 4 row-groups x 2 col-groups; each wave owns 32x64 (2x4 tiles).
// ---------------------------------------------------------------------------
#define TM 128
#define TN 128
#define TK 16
#define ASTR 20   // padded row stride for A tile (conflict-free, 16B aligned)
#define BSTR 132  // padded row stride for B tile

__global__ __launch_bounds__(256) void gemm_wmma_f32(
    const float* __restrict__ A, const float* __restrict__ Bm,
    const float* __restrict__ bias, const float* __restrict__ attnp,
    float* __restrict__ C, float* __restrict__ ssrc, float* __restrict__ sdst,
    int M, int K, int Nc) {
  __shared__ float As[2][TM * ASTR];
  __shared__ float Bs[2][TK * BSTR];

  const int tid = threadIdx.x;
  const int lane = tid & 31;
  const int wave = tid >> 5;
  const int wm = (wave >> 1) * 32;  // wave row offset: 0,32,64,96
  const int wn = (wave & 1) * 64;   // wave col offset: 0,64
  const int blockM = blockIdx.y * TM;
  const int blockN = blockIdx.x * TN;
  const bool fullM = (blockM + TM) <= M;

  v8f zero8 = {0.f, 0.f, 0.f, 0.f, 0.f, 0.f, 0.f, 0.f};
  v8f acc[2][4];
#pragma unroll
  for (int i = 0; i < 2; ++i)
#pragma unroll
    for (int j = 0; j < 4; ++j) acc[i][j] = zero8;

  auto stageA = [&](int kb, int buf) {
#if USE_ASYNC_LDS
    if (fullM) {
      for (int i = tid; i < (TM * TK) / 4; i += 256) {
        int row = i >> 2;
        int c4 = (i & 3) * 4;
        async_copy16(A + (size_t)(blockM + row) * K + kb + c4,
                     &As[buf][row * ASTR + c4]);
      }
      return;
    }
#endif
    for (int i = tid; i < (TM * TK) / 4; i += 256) {
      int row = i >> 2;
      int c4 = (i & 3) * 4;
      int gr = blockM + row;
      float4 v = make_float4(0.f, 0.f, 0.f, 0.f);
      if (gr < M) {
        const float* gp = A + (size_t)gr * K + kb + c4;
        v = *(const float4*)gp;
        if (kb + 2 * TK < K) __builtin_prefetch(gp + 2 * TK, 0, 0);
      }
      *(float4*)(&As[buf][row * ASTR + c4]) = v;
    }
  };
  auto stageB = [&](int kb, int buf) {
    // 16x128 floats = 512 float4, 256 threads -> 2 each
    for (int i = tid; i < (TK * TN) / 4; i += 256) {
      int row = i >> 5;
      int c4 = (i & 31) * 4;
      const float* gp = Bm + (size_t)(kb + row) * Nc + blockN + c4;
#if USE_ASYNC_LDS
      async_copy16(gp, &Bs[buf][row * BSTR + c4]);
#else
      *(float4*)(&Bs[buf][row * BSTR + c4]) = *(const float4*)gp;
#endif
    }
  };

  stageA(0, 0);
  stageB(0, 0);
  async_wait0();
  __syncthreads();

  const int ml = lane & 15;        // A row / B col within a 16-tile
  const int kh = (lane >> 4) * 2;  // K pair select (0 or 2)

  const int nsteps = K / TK;
  for (int s = 0; s < nsteps; ++s) {
    int buf = s & 1;
    if (s + 1 < nsteps) {
      stageA((s + 1) * TK, buf ^ 1);
      stageB((s + 1) * TK, buf ^ 1);
    }
#pragma unroll
    for (int k4 = 0; k4 < TK; k4 += 4) {
      v2f af[2];
      v2f bf[4];
#pragma unroll
      for (int i = 0; i < 2; ++i) {
        const float* ap = &As[buf][(wm + i * 16 + ml) * ASTR + k4 + kh];
        af[i] = *(const v2f*)ap;
      }
#pragma unroll
      for (int j = 0; j < 4; ++j) {
        const float* bp = &Bs[buf][(k4 + kh) * BSTR + wn + j * 16 + ml];
        v2f b;
        b.x = bp[0];
        b.y = bp[BSTR];
        bf[j] = b;
      }
#pragma unroll
      for (int i = 0; i < 2; ++i)
#pragma unroll
        for (int j = 0; j < 4; ++j)
          acc[i][j] = wmma_f32(af[i], bf[j], acc[i][j]);
    }
    async_wait0();
    __syncthreads();
  }

  // ---- store C.  C/D layout: n = lane&15, m = 8*(lane>>4) + vgpr index ----
  const int mh = (lane >> 4) * 8;
#pragma unroll
  for (int i = 0; i < 2; ++i)
#pragma unroll
    for (int j = 0; j < 4; ++j) {
      int gn = blockN + wn + j * 16 + ml;
      float bb = bias ? bias[gn] : 0.f;
#pragma unroll
      for (int v = 0; v < 8; ++v) {
        int gm = blockM + wm + i * 16 + mh + v;
        if (gm < M) C[(size_t)gm * Nc + gn] = acc[i][j][v] + bb;
      }
    }

  // ---- fused attention dot-products (s_src / s_dst partials) ----
  if (attnp) {
    float a0[4], a1[4];
#pragma unroll
    for (int j = 0; j < 4; ++j) {
      int gn = blockN + wn + j * 16 + ml;
      a0[j] = attnp[gn];
      a1[j] = attnp[Nc + gn];
    }
#pragma unroll
    for (int i = 0; i < 2; ++i) {
#pragma unroll
      for (int v = 0; v < 8; ++v) {
        float p0 = 0.f, p1 = 0.f;
#pragma unroll
        for (int j = 0; j < 4; ++j) {
          p0 += acc[i][j][v] * a0[j];
          p1 += acc[i][j][v] * a1[j];
        }
        // reduce across the 16 lanes of each half-wave
        for (int off = 8; off > 0; off >>= 1) {
          p0 += __shfl_xor(p0, off, 16);
          p1 += __shfl_xor(p1, off, 16);
        }
        int gm = blockM + wm + i * 16 + mh + v;
        if (ml == 0 && gm < M) {
          unsafeAtomicAdd(&ssrc[gm], p0);
          unsafeAtomicAdd(&sdst[gm], p1);
        }
      }
    }
  }
}

// ---------------------------------------------------------------------------
// Small helper kernels
// ---------------------------------------------------------------------------
__global__ void transpose_k(const float* __restrict__ w, float* __restrict__ wt,
                            int D) {
  int i = blockIdx.x * blockDim.x + threadIdx.x;
  if (i >= D * D) return;
  int k = i / D, o = i - k * D;
  wt[i] = w[(size_t)o * D + k];  // wt[k][o] = w[o][k]
}

__global__ void copy_k(const float* __restrict__ src, float* __restrict__ dst,
                       int n) {
  int i = blockIdx.x * blockDim.x + threadIdx.x;
  if (i < n) dst[i] = src[i];
}

__global__ void zero_k(float* __restrict__ p, int n) {
  int i = blockIdx.x * blockDim.x + threadIdx.x;
  if (i < n) p[i] = 0.f;
}

__global__ void relu_k(const float* __restrict__ in, float* __restrict__ out,
                       int n) {
  int i = blockIdx.x * blockDim.x + threadIdx.x;
  if (i < n) out[i] = fmaxf(in[i], 0.f);
}

// acc[n*D+d] = x[n*D+d] + bias[d]   (residual + bias init)
__global__ void initacc_k(const float* __restrict__ x,
                          const float* __restrict__ bias,
                          float* __restrict__ acc, int D, int total) {
  int i = blockIdx.x * blockDim.x + threadIdx.x;
  if (i < total) acc[i] = x[i] + bias[i % D];
}

// wts[r,i,o] = sum_b att[r,b] * basis[b,i,o]
__global__ void combine_k(const float* __restrict__ att,
                          const float* __restrict__ basis,
                          float* __restrict__ wts, int Bn, int DD, int total) {
  int i = blockIdx.x * blockDim.x + threadIdx.x;
  if (i >= total) return;
  int r = i / DD;
  int io = i - r * DD;
  float s = 0.f;
  for (int b = 0; b < Bn; ++b) s += att[r * Bn + b] * basis[(size_t)b * DD + io];
  wts[i] = s;
}

// pass 1: e = exp(-leaky_relu(s_src[src]+s_dst[dst])); deg[src]+=1; asum[src]+=e
__global__ void edge_pass1(const int* __restrict__ triples,
                           const int* __restrict__ cp,
                           const float* __restrict__ ssrc,
                           const float* __restrict__ sdst,
                           float* __restrict__ deg, float* __restrict__ asum,
                           float* __restrict__ ev, int r) {
  int e0 = cp[r], e1 = cp[r + 1];
  for (int e = e0 + blockIdx.x * blockDim.x + threadIdx.x; e < e1;
       e += gridDim.x * blockDim.x) {
    int s = triples[3 * e + 0];
    int d = triples[3 * e + 2];
    float v = ssrc[s] + sdst[d];
    float lr = v > 0.f ? v : NEG_SLOPE * v;
    float ex = expf(-lr);
    ev[e] = ex;
    unsafeAtomicAdd(&deg[s], 1.f);
    unsafeAtomicAdd(&asum[s], ex);
  }
}

// pass 2: c = e / max(asum[dst],eps) / max(deg[src],eps);
//         acc[src,:] += c * Wh[dst,:]        (one wave per edge)
__global__ __launch_bounds__(256) void edge_pass2(
    const int* __restrict__ triples, const int* __restrict__ cp,
    const float* __restrict__ ev, const float* __restrict__ deg,
    const float* __restrict__ asum, const float* __restrict__ wh,
    float* __restrict__ acc, int r, int D) {
  int e0 = cp[r], e1 = cp[r + 1];
  int gwave = (blockIdx.x * blockDim.x + threadIdx.x) >> 5;
  int lane = threadIdx.x & 31;
  int totalWaves = (gridDim.x * blockDim.x) >> 5;
  for (int e = e0 + gwave; e < e1; e += totalWaves) {
    int s = triples[3 * e + 0];
    int d = triples[3 * e + 2];
    float c = ev[e] / fmaxf(asum[d], 1e-6f) / fmaxf(deg[s], 1e-6f);
    const float4* wp = (const float4*)(wh + (size_t)d * D);
    float* hp = acc + (size_t)s * D;
    for (int c4 = lane; c4 * 4 < D; c4 += 32) {
      float4 w = wp[c4];
      unsafeAtomicAdd(hp + 4 * c4 + 0, c * w.x);
      unsafeAtomicAdd(hp + 4 * c4 + 1, c * w.y);
      unsafeAtomicAdd(hp + 4 * c4 + 2, c * w.z);
      unsafeAtomicAdd(hp + 4 * c4 + 3, c * w.w);
    }
  }
}

// ---------------------------------------------------------------------------
// Host orchestration
// ---------------------------------------------------------------------------
extern "C" void kernel_launch(void* const* d_in, const int* in_sizes, int n_in,
                              void* d_out, int out_size, void* d_ws,
                              size_t ws_size, hipStream_t stream) {
  const float* feats = (const float*)d_in[0];
  const float* emb = (const float*)d_in[1];
  const float* lin_w = (const float*)d_in[2];
  const float* lin_b = (const float*)d_in[3];
  const float* basis[2] = {(const float*)d_in[4], (const float*)d_in[8]};
  const float* att[2] = {(const float*)d_in[5], (const float*)d_in[9]};
  const float* attn[2] = {(const float*)d_in[6], (const float*)d_in[10]};
  const float* bias[2] = {(const float*)d_in[7], (const float*)d_in[11]};
  const int* triples = (const int*)d_in[12];
  const int* cp = (const int*)d_in[13];

  const int D = in_sizes[3];
  const int NT_A = in_sizes[0] / D;
  const int NT_B = in_sizes[1] / D;
  const int N = NT_A + NT_B;
  const int Bn = in_sizes[4] / (D * D);
  const int R = in_sizes[13] - 1;

  // workspace layout (floats)
  float* X = (float*)d_ws;               // N*D   current layer input
  float* H = X + (size_t)N * D;          // N*D   layer-0 accumulator
  float* WH = H + (size_t)N * D;         // N*D   per-relation Wh
  float* WTS = WH + (size_t)N * D;       // R*D*D combined weights (per layer)
  float* LWT = WTS + (size_t)R * D * D;  // D*D   lin_w^T
  float* SSRC = LWT + (size_t)D * D;     // N  -+
  float* SDST = SSRC + N;                // N   | contiguous: zeroed as one
  float* DEG = SDST + N;                 // N   | block of 4N floats
  float* ASUM = DEG + N;                 // N  -+
  float* EV = ASUM + N;                  // E
  float* OUT = (float*)d_out;

  const int ND = N * D;
  const int DD = D * D;

  // input projection: X[0:NT_A] = feats @ lin_w^T + lin_b ; X[NT_A:] = emb
  transpose_k<<<(DD + 255) / 256, 256, 0, stream>>>(lin_w, LWT, D);
  {
    dim3 grid(D / TN, (NT_A + TM - 1) / TM);
    gemm_wmma_f32<<<grid, 256, 0, stream>>>(feats, LWT, lin_b, nullptr, X,
                                            nullptr, nullptr, NT_A, D, D);
  }
  copy_k<<<(NT_B * D + 255) / 256, 256, 0, stream>>>(emb, X + (size_t)NT_A * D,
                                                     NT_B * D);

  for (int L = 0; L < 2; ++L) {
    float* ACC = (L == 0) ? H : OUT;
    combine_k<<<(R * DD + 255) / 256, 256, 0, stream>>>(att[L], basis[L], WTS,
                                                        Bn, DD, R * DD);
    initacc_k<<<(ND + 255) / 256, 256, 0, stream>>>(X, bias[L], ACC, D, ND);
    for (int r = 0; r < R; ++r) {
      // zero SSRC,SDST,DEG,ASUM (contiguous 4N block)
      zero_k<<<(4 * N + 255) / 256, 256, 0, stream>>>(SSRC, 4 * N);
      dim3 grid(D / TN, (N + TM - 1) / TM);
      gemm_wmma_f32<<<grid, 256, 0, stream>>>(
          X, WTS + (size_t)r * DD, nullptr, attn[L] + (size_t)r * 2 * D, WH,
          SSRC, SDST, N, D, D);
      edge_pass1<<<1024, 256, 0, stream>>>(triples, cp, SSRC, SDST, DEG, ASUM,
                                           EV, r);
      edge_pass2<<<2048, 256, 0, stream>>>(triples, cp, EV, DEG, ASUM, WH, ACC,
                                           r, D);
    }
    if (L == 0) relu_k<<<(ND + 255) / 256, 256, 0, stream>>>(H, X, ND);
  }
}